// PhaseNet_15229954031692
// MI455X (gfx1250) — compile-verified
//
#include <hip/hip_runtime.h>
#include <math.h>

// ============================================================================
// PhaseNet on MI455X (gfx1250), fp32 end-to-end.
//
// phase_layer() is linear in its input; with u_k = Re X_k, v_k = -Im X_k and
// irfft(rfft(h)) == h (jnp irfft ignores Im of bin 0 / Nyquist):
//   phase_layer(h) = out_scale .* ( h + sum_{k<64} p_k cos(tkn) - q_k sin(tkn) )
//   p = alpha.*(ReM - u), q = alpha.*(ImM + v), alpha_0=1/D, alpha_k=2/D
//   ReM_k = sum_h I[h,k](cos(phi_h) u_h + sin(phi_h) v_h)
//   ImM_k = sum_h I[h,k](sin(phi_h) u_h - cos(phi_h) v_h)
// => rank-128 update: UV = Hn @ G^T (G built N-major), Ycorr = UV @ MixH^T
// (MixH built N-major). Three fp32 WMMA GEMMs total (the third is lm_head,
// which dominates: 823 MB logits write + 206 MB weights ~= 45 us at 23.3 TB/s,
// so the problem is DRAM-bound and fp32 WMMA loses nothing vs bf16).
// ============================================================================

typedef float v2f __attribute__((ext_vector_type(2)));
typedef float v8f __attribute__((ext_vector_type(8)));

#define D_DIM 1024
#define H_DIM 64
#define NROWS 4096            // B*S = 2*2048
#define SEQ   2048
#define VOCAB 50257
#define EPSF  1.1920928955078125e-07f
#define TWO_PI 6.28318530717958647692f

// ---------------------------------------------------------------------------
// Embedding gather: x[r,:] = embed[tokens[r],:]
// ---------------------------------------------------------------------------
__global__ void k_embed(const int* __restrict__ tokens,
                        const float* __restrict__ embed,
                        float* __restrict__ x) {
  int r = blockIdx.x;
  int tok = tokens[r];
  const float4* src = (const float4*)(embed + (size_t)tok * D_DIM);
  float4* dst = (float4*)(x + (size_t)r * D_DIM);
  dst[threadIdx.x] = src[threadIdx.x];   // 256 threads * float4 = 1024 floats
}

// ---------------------------------------------------------------------------
// RMSNorm (+ optional affine): out = x * rsqrt(mean(x^2)+eps) * w [* g + b]
// One 256-thread block per row. The row is staged into LDS with the gfx1250
// async-copy path (global_load_async_to_lds_b128 + s_wait_asynccnt).
// ---------------------------------------------------------------------------
__global__ void k_rmsnorm(const float* __restrict__ x,
                          const float* __restrict__ w,
                          const float* __restrict__ gamma,
                          const float* __restrict__ beta,
                          float* __restrict__ out) {
  __shared__ float srow[D_DIM];    // 4 KB row tile
  __shared__ float red[256];
  int r = blockIdx.x, t = threadIdx.x;

  // Async stage: each thread copies its 16-byte slice global -> LDS.
  {
    const float* gsrc = x + (size_t)r * D_DIM + (size_t)t * 4;
    unsigned lds_off = (unsigned)(size_t)&srow[t * 4];
    unsigned long long gaddr = (unsigned long long)(size_t)gsrc;
    asm volatile(
        "global_load_async_to_lds_b128 %0, %1, off\n\t"
        "s_wait_asynccnt 0"
        :
        : "v"(lds_off), "v"(gaddr)
        : "memory");
  }
  __syncthreads();

  float4 v = *(const float4*)&srow[t * 4];
  red[t] = v.x * v.x + v.y * v.y + v.z * v.z + v.w * v.w;
  __syncthreads();
  for (int off = 128; off > 0; off >>= 1) {
    if (t < off) red[t] += red[t + off];
    __syncthreads();
  }
  float scale = rsqrtf(red[0] * (1.0f / (float)D_DIM) + EPSF);
  float4 wv = ((const float4*)w)[t];
  float4 o;
  o.x = v.x * scale * wv.x; o.y = v.y * scale * wv.y;
  o.z = v.z * scale * wv.z; o.w = v.w * scale * wv.w;
  if (gamma != nullptr) {
    float4 g = ((const float4*)gamma)[t];
    float4 b = ((const float4*)beta)[t];
    o.x = o.x * g.x + b.x; o.y = o.y * g.y + b.y;
    o.z = o.z * g.z + b.z; o.w = o.w * g.w + b.w;
  }
  ((float4*)(out + (size_t)r * D_DIM))[t] = o;
}

// ---------------------------------------------------------------------------
// Build G in N-major (N x K = 128 x 1024) layout:
//   G[j, m] = cos(t*j*m) for j<64, sin(t*(j-64)*m) for j>=64
// ---------------------------------------------------------------------------
__global__ void k_build_G(float* __restrict__ G) {
  int idx = blockIdx.x * blockDim.x + threadIdx.x;   // 0..131071
  int j = idx >> 10;          // 0..127  (output/frequency index)
  int m = idx & 1023;         // 0..1023 (k/reduction index)
  int k = j & 63;
  float ang = (TWO_PI / (float)D_DIM) * (float)((m * k) % D_DIM);
  G[idx] = (j < 64) ? cosf(ang) : sinf(ang);
}

// ---------------------------------------------------------------------------
// Build MixH in N-major (N x K = 1024 x 128) layout: MixH[n, j].
// Mix folds phases/interference/alpha into the (cos | -sin) back-projection.
// ---------------------------------------------------------------------------
__global__ void k_build_MixH(const float* __restrict__ phase,  // (64)
                             const float* __restrict__ inter,  // (64,64) [h*64+k]
                             float* __restrict__ MixH) {
  int idx = blockIdx.x * blockDim.x + threadIdx.x;   // 0..131071
  int n = idx >> 7;           // 0..1023 (output channel)
  int j = idx & 127;          // 0..127  (uv reduction index)
  int h = j & 63;
  bool upart = (j < 64);
  float sph, cph;
  sincosf(phase[h], &sph, &cph);
  float acc = 0.0f;
  for (int k = 0; k < 64; ++k) {
    float alpha = (k == 0) ? (1.0f / (float)D_DIM) : (2.0f / (float)D_DIM);
    float Ihk = inter[h * 64 + k];
    float mp, mq;   // contributions of uv[j] to p_k and q_k
    if (upart) {
      mp = alpha * (Ihk * cph - ((h == k) ? 1.0f : 0.0f));
      mq = alpha * (Ihk * sph);
    } else {
      mp = alpha * (Ihk * sph);
      mq = alpha * (((h == k) ? 1.0f : 0.0f) - Ihk * cph);
    }
    float ang = (TWO_PI / (float)D_DIM) * (float)((k * n) % D_DIM);
    float sn, cn;
    sincosf(ang, &sn, &cn);
    acc += mp * cn - mq * sn;   // y_corr = sum_k p_k cos - q_k sin
  }
  MixH[idx] = acc;
}

// ---------------------------------------------------------------------------
// fp32 WMMA GEMM: C(MxN) = A(MxK, row-major) @ B^T, B stored N x K row-major
// (B[n*K + k]) so all fragment loads are contiguous float2 (global_load_b64).
//
// Block: 256 threads = 8 waves; 64(M) x 128(N) tile per block.
// Wave w: M-subtile (w&3)*16; four N-subtiles (w>>2)*64 + {0,16,32,48}
// (A fragment loaded once, reused for 4 WMMAs).
//
// Inner loop is software-pipelined at source level: K is consumed in chunks
// of 16; all 20 fragment loads of a chunk are issued before its 16 WMMAs so
// the wave retires WMMAs against a descending s_wait_loadcnt ladder instead
// of stalling on loadcnt==0 per WMMA.
//
// V_WMMA_F32_16X16X4_F32 fragment layout (ISA 7.12.2):
//   A 16x4: lanes 0-15 hold K={0,1}, lanes 16-31 hold K={2,3} (2 VGPRs)
//   B 4x16: mirrored; C/D 16x16 in 8 VGPRs (lanes 0-15 rows i, 16-31 rows 8+i).
// ---------------------------------------------------------------------------
__global__ void k_wmma_gemm_nk(const float* __restrict__ A,
                               const float* __restrict__ B,
                               float* __restrict__ C,
                               int M, int N, int K) {
  int lane  = threadIdx.x & 31;
  int wave  = threadIdx.x >> 5;
  int l15   = lane & 15;
  int khalf = (lane >> 4) << 1;                 // 0 (lanes 0-15) or 2 (16-31)

  int m0 = blockIdx.y * 64 + (wave & 3) * 16;
  int n0 = blockIdx.x * 128 + (wave >> 2) * 64;

  const float* Ap = A + (size_t)(m0 + l15) * K + khalf;

  const float* Bp[4];
#pragma unroll
  for (int j = 0; j < 4; ++j) {
    int c = n0 + 16 * j + l15;
    if (c > N - 1) c = N - 1;                   // clamp: no EXEC divergence
    Bp[j] = B + (size_t)c * K + khalf;
  }

  v8f acc[4];
#pragma unroll
  for (int j = 0; j < 4; ++j)
    acc[j] = (v8f){0.f, 0.f, 0.f, 0.f, 0.f, 0.f, 0.f, 0.f};

  for (int k = 0; k < K; k += 16) {             // chunk = 4 k-steps of 4
    v2f a[4], b[4][4];
    // ---- issue all 20 loads of the chunk first ----
#pragma unroll
    for (int u = 0; u < 4; ++u)
      a[u] = *(const v2f*)(Ap + k + 4 * u);
#pragma unroll
    for (int u = 0; u < 4; ++u)
#pragma unroll
      for (int j = 0; j < 4; ++j)
        b[u][j] = *(const v2f*)(Bp[j] + k + 4 * u);
    // ---- then 16 WMMAs ----
#pragma unroll
    for (int u = 0; u < 4; ++u)
#pragma unroll
      for (int j = 0; j < 4; ++j)
        acc[j] = __builtin_amdgcn_wmma_f32_16x16x4_f32(false, a[u], false,
                                                       b[u][j], (short)0,
                                                       acc[j], false, false);
  }

  int rbase = m0 + ((lane >> 4) << 3);
#pragma unroll
  for (int i = 0; i < 8; ++i) {
    size_t ro = (size_t)(rbase + i) * (size_t)N;
#pragma unroll
    for (int j = 0; j < 4; ++j) {
      int col = n0 + 16 * j + l15;
      if (col < N) C[ro + col] = acc[j][i];
    }
  }
}

// ---------------------------------------------------------------------------
// Fuse: x += 0.3 * silu( (hn + ycorr)*out_scale  +  dwconv3_seq(hn) )
// conv along sequence axis (zero pad, per-channel kernel of 3), per batch.
// ---------------------------------------------------------------------------
__global__ void k_fuse(float* __restrict__ x,
                       const float* __restrict__ hn,
                       const float* __restrict__ yc,
                       const float* __restrict__ oscale,
                       const float* __restrict__ ck) {   // (D,1,3) -> ck[d*3+t]
  int r = blockIdx.x;
  int s = r & (SEQ - 1);
  size_t base = (size_t)r * D_DIM;
  for (int d = threadIdx.x; d < D_DIM; d += blockDim.x) {
    float hc = hn[base + d];
    float hm = (s > 0)       ? hn[base - D_DIM + d] : 0.0f;
    float hp = (s < SEQ - 1) ? hn[base + D_DIM + d] : 0.0f;
    float conv = ck[d * 3 + 0] * hm + ck[d * 3 + 1] * hc + ck[d * 3 + 2] * hp;
    float ph = (hc + yc[base + d]) * oscale[d];
    float z = ph + conv;
    float silu = z / (1.0f + expf(-z));
    x[base + d] += 0.3f * silu;
  }
}

// ---------------------------------------------------------------------------
// Host launcher
// ---------------------------------------------------------------------------
extern "C" void kernel_launch(void* const* d_in, const int* in_sizes, int n_in,
                              void* d_out, int out_size, void* d_ws, size_t ws_size,
                              hipStream_t stream) {
  const int*   tokens = (const int*)  d_in[0];
  const float* embed  = (const float*)d_in[1];
  const float* phase  = (const float*)d_in[2];
  const float* inter  = (const float*)d_in[3];
  const float* oscale = (const float*)d_in[4];
  const float* gammas = (const float*)d_in[5];
  const float* betas  = (const float*)d_in[6];
  const float* nws    = (const float*)d_in[7];
  const float* convk  = (const float*)d_in[8];
  const float* onw    = (const float*)d_in[9];
  const float* lmw    = (const float*)d_in[10];
  float* out = (float*)d_out;

  // Workspace layout (floats): ~53.5 MB total
  float* ws   = (float*)d_ws;
  float* x    = ws;                                   // 4096*1024
  float* hn   = x    + (size_t)NROWS * D_DIM;         // 4096*1024
  float* yc   = hn   + (size_t)NROWS * D_DIM;         // 4096*1024
  float* uv   = yc   + (size_t)NROWS * D_DIM;         // 4096*128
  float* G    = uv   + (size_t)NROWS * 128;           // 128*1024 (N-major)
  float* MixH = G    + (size_t)128 * D_DIM;           // 1024*128 (N-major)

  // One-time (cheap, deterministic) basis/mixing tables
  k_build_G   <<<512, 256, 0, stream>>>(G);
  k_build_MixH<<<512, 256, 0, stream>>>(phase, inter, MixH);

  // Embedding
  k_embed<<<NROWS, 256, 0, stream>>>(tokens, embed, x);

  // 4 layers
  for (int c = 0; c < 4; ++c) {
    k_rmsnorm<<<NROWS, 256, 0, stream>>>(x, nws + c * D_DIM,
                                         gammas + c * D_DIM, betas + c * D_DIM, hn);
    // UV = Hn @ G^T        (4096x1024 @ [128x1024]^T)
    k_wmma_gemm_nk<<<dim3(1, NROWS / 64), 256, 0, stream>>>(hn, G, uv,
                                                            NROWS, 128, D_DIM);
    // Ycorr = UV @ MixH^T  (4096x128 @ [1024x128]^T)
    k_wmma_gemm_nk<<<dim3(D_DIM / 128, NROWS / 64), 256, 0, stream>>>(
        uv, MixH, yc, NROWS, D_DIM, 128);
    k_fuse<<<NROWS, 256, 0, stream>>>(x, hn, yc, oscale, convk);
  }

  // Final norm (no affine) + lm_head: logits = Xf @ W^T, W is (V,D) = NxK
  k_rmsnorm<<<NROWS, 256, 0, stream>>>(x, onw, nullptr, nullptr, hn);
  k_wmma_gemm_nk<<<dim3((VOCAB + 127) / 128, NROWS / 64), 256, 0, stream>>>(
      hn, lmw, out, NROWS, VOCAB, D_DIM);
}